// FeatherNet_83038897701629
// MI455X (gfx1250) — compile-verified
//
#include <hip/hip_runtime.h>
#include <hip/hip_bf16.h>

typedef __attribute__((ext_vector_type(2))) float v2f;
typedef __attribute__((ext_vector_type(8))) float v8f;

namespace {

constexpr int D_IN  = 2048;
constexpr int D_FF  = 8192;
constexpr int D_OUT = 2048;
constexpr long long N_W1 = (long long)D_FF * D_IN;   // 16777216
constexpr int       N_B1 = D_FF;                     // 8192
constexpr long long N_W2 = (long long)D_OUT * D_FF;  // 16777216
constexpr int SIZE_N = 5794;
constexpr int SIZE_M = 2897;
constexpr int BATCH  = 4096;

constexpr int BLOCK_M = 128;   // 2 waves in M
constexpr int BLOCK_N = 256;   // 4 waves in N
// per-wave tile: 64x64 = 4x4 grid of 16x16 WMMA tiles

__device__ __forceinline__ v8f wmma_f32(v2f a, v2f b, v8f c) {
  // D(16x16 f32) = A(16x4 f32) x B(4x16 f32) + C
  // 8 args: (neg_a, A, neg_b, B, c_mod, C, reuse_a, reuse_b)
  return __builtin_amdgcn_wmma_f32_16x16x4_f32(
      false, a, false, b, (short)0, c, false, false);
}

// NT GEMM main loop: C_tile += A[m, :] . B[n, :]  (dot over K, both row-major)
// A-lane layout (16x4 f32): lane<16 -> row m_base+lane, k=k0..k0+1
//                           lane>=16 -> row m_base+lane-16, k=k0+2..k0+3
// B-lane layout (4x16 f32) is the symmetric transpose-feed: same addressing.
template<bool GUARD>
__device__ __forceinline__ void gemm_body(
    const float* __restrict__ A, const float* __restrict__ B,
    int M, int N, int K, int lda, int ldb,
    int mw, int nw, int lane, v8f acc[4][4]) {
  const int lr = lane & 15;
  const int kq = (lane >> 4) << 1;   // 0 or 2

  const float* ap[4];
  const float* bp[4];
  float amask[4], bmask[4];
#pragma unroll
  for (int i = 0; i < 4; ++i) {
    int ar = mw + i * 16 + lr;
    int br = nw + i * 16 + lr;
    amask[i] = (!GUARD || ar < M) ? 1.0f : 0.0f;
    bmask[i] = (!GUARD || br < N) ? 1.0f : 0.0f;
    if (GUARD) {
      ar = (ar < M) ? ar : (M - 1);
      br = (br < N) ? br : (N - 1);
    }
    ap[i] = A + (size_t)ar * lda + kq;
    bp[i] = B + (size_t)br * ldb + kq;
  }

  const int kmain = K & ~3;
  for (int k = 0; k < kmain; k += 4) {
    v2f a[4], b[4];
#pragma unroll
    for (int i = 0; i < 4; ++i) a[i] = *(const v2f*)(ap[i] + k);
#pragma unroll
    for (int i = 0; i < 4; ++i) b[i] = *(const v2f*)(bp[i] + k);
    if (GUARD) {
#pragma unroll
      for (int i = 0; i < 4; ++i) { a[i] *= amask[i]; b[i] *= bmask[i]; }
    }
#pragma unroll
    for (int mi = 0; mi < 4; ++mi)
#pragma unroll
      for (int ni = 0; ni < 4; ++ni)
        acc[mi][ni] = wmma_f32(a[mi], b[ni], acc[mi][ni]);
  }

  if (K & 3) {  // ragged K tail (e.g. K=2897): zero-fill OOB k
    const int k = kmain;
    v2f a[4], b[4];
#pragma unroll
    for (int i = 0; i < 4; ++i) {
      a[i].x = (k + kq + 0 < K) ? ap[i][k + 0] : 0.0f;
      a[i].y = (k + kq + 1 < K) ? ap[i][k + 1] : 0.0f;
      b[i].x = (k + kq + 0 < K) ? bp[i][k + 0] : 0.0f;
      b[i].y = (k + kq + 1 < K) ? bp[i][k + 1] : 0.0f;
      if (GUARD) { a[i] *= amask[i]; b[i] *= bmask[i]; }
    }
#pragma unroll
    for (int mi = 0; mi < 4; ++mi)
#pragma unroll
      for (int ni = 0; ni < 4; ++ni)
        acc[mi][ni] = wmma_f32(a[mi], b[ni], acc[mi][ni]);
  }
}

// C[M,N] = op( A[M,K] @ B[N,K]^T + bias[N] ), row-major everything.
template<bool BIAS, bool RELU>
__global__ __launch_bounds__(256)
void gemm_nt_kernel(const float* __restrict__ A, const float* __restrict__ B,
                    const float* __restrict__ bias, float* __restrict__ C,
                    int M, int N, int K, int lda, int ldb, int ldc) {
  const int lane = threadIdx.x & 31;
  const int wave = threadIdx.x >> 5;     // 8 waves
  const int wm = wave >> 2;              // 0..1
  const int wn = wave & 3;               // 0..3
  const int m0 = blockIdx.y * BLOCK_M;
  const int n0 = blockIdx.x * BLOCK_N;
  const int mw = m0 + wm * 64;
  const int nw = n0 + wn * 64;

  v8f acc[4][4];
  const v8f zacc = {};
#pragma unroll
  for (int mi = 0; mi < 4; ++mi)
#pragma unroll
    for (int ni = 0; ni < 4; ++ni) acc[mi][ni] = zacc;

  const bool interior = (m0 + BLOCK_M <= M) && (n0 + BLOCK_N <= N);
  if (interior) gemm_body<false>(A, B, M, N, K, lda, ldb, mw, nw, lane, acc);
  else          gemm_body<true >(A, B, M, N, K, lda, ldb, mw, nw, lane, acc);

  // Epilogue. C/D layout: VGPR v, lane<16 -> (M = v,   N = lane)
  //                               lane>=16 -> (M = v+8, N = lane-16)
  const int lr = lane & 15;
  const int rh = (lane >> 4) * 8;
#pragma unroll
  for (int ni = 0; ni < 4; ++ni) {
    const int col = nw + ni * 16 + lr;
    if (col >= N) continue;
    const float bv = BIAS ? bias[col] : 0.0f;
#pragma unroll
    for (int mi = 0; mi < 4; ++mi) {
#pragma unroll
      for (int v = 0; v < 8; ++v) {
        const int row = mw + mi * 16 + rh + v;
        if (row < M) {
          float val = acc[mi][ni][v] + bv;
          if (RELU) val = fmaxf(val, 0.0f);
          C[(size_t)row * ldc + col] = val;
        }
      }
    }
  }
}

}  // namespace

extern "C" void kernel_launch(void* const* d_in, const int* in_sizes, int n_in,
                              void* d_out, int out_size, void* d_ws, size_t ws_size,
                              hipStream_t stream) {
  (void)in_sizes; (void)n_in; (void)out_size; (void)ws_size;
  const float* x  = (const float*)d_in[0];   // [4096, 2048]
  const float* V1 = (const float*)d_in[1];   // [5794, 2897]
  const float* V2 = (const float*)d_in[2];   // [5794, 2897]
  float* out = (float*)d_out;                // [4096, 2048]

  float* ws = (float*)d_ws;
  const size_t vflat_elems = ((size_t)SIZE_N * SIZE_N + 255) & ~(size_t)255;
  float* Vf = ws;                    // V = V1 @ V2^T, flattened [5794*5794]
  float* h  = ws + vflat_elems;      // [4096, 8192]

  const dim3 blk(256);

  // GEMM 1: V[i,j] = sum_k V1[i,k] * V2[j,k]  (M=N=5794, K=2897)
  {
    dim3 grd((SIZE_N + BLOCK_N - 1) / BLOCK_N, (SIZE_N + BLOCK_M - 1) / BLOCK_M);
    gemm_nt_kernel<false, false><<<grd, blk, 0, stream>>>(
        V1, V2, nullptr, Vf, SIZE_N, SIZE_N, SIZE_M, SIZE_M, SIZE_M, SIZE_N);
  }

  const float* W1 = Vf;                              // [8192, 2048]
  const float* b1 = Vf + N_W1;                       // [8192]
  const float* W2 = Vf + N_W1 + N_B1;                // [2048, 8192]
  const float* b2 = Vf + N_W1 + N_B1 + N_W2;         // [2048]

  // GEMM 2: h = relu(x @ W1^T + b1)  (M=4096, N=8192, K=2048)
  {
    dim3 grd(D_FF / BLOCK_N, BATCH / BLOCK_M);
    gemm_nt_kernel<true, true><<<grd, blk, 0, stream>>>(
        x, W1, b1, h, BATCH, D_FF, D_IN, D_IN, D_IN, D_FF);
  }

  // GEMM 3: out = h @ W2^T + b2  (M=4096, N=2048, K=8192)
  {
    dim3 grd(D_OUT / BLOCK_N, BATCH / BLOCK_M);
    gemm_nt_kernel<true, false><<<grd, blk, 0, stream>>>(
        h, W2, b2, out, BATCH, D_OUT, D_FF, D_FF, D_FF, D_OUT);
  }
}